// NodeUpdateNetFG_5059471474799
// MI455X (gfx1250) — compile-verified
//
#include <hip/hip_runtime.h>

typedef __attribute__((ext_vector_type(2))) float v2f;
typedef __attribute__((ext_vector_type(8))) float v8f;
typedef int v4i __attribute__((vector_size(16)));

#define LN_EPS 1e-5f
#define SLOPE 0.01f
#define TILES_A 8

// CDNA5 async global->LDS path (ASYNCcnt). Guarded so a missing builtin
// degrades to a sync fallback instead of breaking the build.
#if defined(__gfx1250__) && \
    __has_builtin(__builtin_amdgcn_global_load_async_to_lds_b128) && \
    __has_builtin(__builtin_amdgcn_s_wait_asynccnt)
#define USE_ASYNC 1
#else
#define USE_ASYNC 0
#endif

__device__ __forceinline__ float leaky(float v) {
  return v > 0.0f ? v : SLOPE * v;
}

__device__ __forceinline__ void cp16_async(const void* g, void* l) {
#if USE_ASYNC
  __builtin_amdgcn_global_load_async_to_lds_b128(
      (__attribute__((address_space(1))) v4i*)g,
      (__attribute__((address_space(3))) v4i*)l, 0, 0);
#else
  *(float4*)l = *(const float4*)g;
#endif
}

template <int N>
__device__ __forceinline__ void wait_async_le() {
#if USE_ASYNC
  __builtin_amdgcn_s_wait_asynccnt(N);   // immediate required
#endif
}

// ---------------------------------------------------------------- zero ws
__global__ void k_zero(float* __restrict__ p, int n) {
  int i = blockIdx.x * blockDim.x + threadIdx.x;
  if (i < n) p[i] = 0.0f;
}

// ---------------------------------------------------------------- edge MLP + scatter
// 256 threads = 8 wave32; wave w owns output-feature tile [16w,16w+16).
// Per 16-edge tile: C(16x16) = A(16x192) x W1(192x16) via 48 v_wmma_f32_16x16x4_f32.
// A tiles are double-buffered in LDS and staged with async b128 copies so the
// gather of tile t+1 overlaps the WMMA chain of tile t.
__global__ __launch_bounds__(256) void k_edge(
    const float* __restrict__ x, const long long* __restrict__ ei,
    const float* __restrict__ ea, const float* __restrict__ W1,
    const float* __restrict__ b1, const float* __restrict__ g1,
    const float* __restrict__ be1, float* __restrict__ sums,
    float* __restrict__ counts, int E, long long Eoff, int tiles_total) {
  __shared__ float As[2][16 * 196];     // 16 edges x 192 (pad->196: conflict-free, 16B-aligned rows)
  __shared__ int   scol2[2][16];
  __shared__ float red_s[8 * 16];
  __shared__ float red_q[8 * 16];
  __shared__ float smean[16], srstd[16];

  const int tid  = threadIdx.x;
  const int lane = tid & 31;
  const int wv   = tid >> 5;
  const int l15  = lane & 15;
  const int hi   = lane >> 4;   // half-wave: 0 or 1
  const int kh   = hi << 1;     // K phase 0 or 2 (A/B operand layout)
  const int gn   = (wv << 4) + l15;

  int nt = tiles_total - blockIdx.x * TILES_A;
  if (nt <= 0) return;
  if (nt > TILES_A) nt = TILES_A;

  const float bias = b1[gn];
  const float gam  = g1[gn];
  const float bet  = be1[gn];
  const float* wp  = W1 + kh * 128 + gn;   // B: lane = col N, vgpr pair = K rows

  // Stage tile t into buffer buf: 3 async b128 per thread (2 x-gather + 1 edge_attr).
  auto stage = [&](int t, int buf) {
    long long base = ((long long)blockIdx.x * TILES_A + t) * 16;
    #pragma unroll
    for (int k = 0; k < 2; ++k) {
      int i = tid + k * 256;              // 512 x-fragments
      int m = i >> 5, c4 = i & 31;
      long long e = base + m;
      long long row = (e < E) ? (long long)ei[e] : 0;
      cp16_async(x + row * 128 + c4 * 4, &As[buf][m * 196 + c4 * 4]);
    }
    {
      int m = tid >> 4, c4 = tid & 15;    // 256 edge_attr fragments
      long long e = base + m;
      long long ec = (e < E) ? e : 0;
      cp16_async(ea + ec * 64 + c4 * 4, &As[buf][m * 196 + 128 + c4 * 4]);
    }
    if (tid < 16) {
      long long e = base + tid;
      scol2[buf][tid] = (e < E) ? (int)ei[Eoff + e] : -1;
    }
  };

  stage(0, 0);
  for (int t = 0; t < nt; ++t) {
    const int cur = t & 1;
    if (t + 1 < nt) {
      stage(t + 1, cur ^ 1);             // prefetch next tile (3 more async/thread)
      wait_async_le<3>();                // retire current buffer's 3 copies (in-order)
    } else {
      wait_async_le<0>();
    }
    __syncthreads();

    // ---- GEMM: 48 f32 WMMAs over K=192
    const float* ap = &As[cur][l15 * 196 + kh];  // A: lane = row M, vgpr pair = K kh..kh+1
    v8f acc = {};
    #pragma unroll 8
    for (int kk = 0; kk < 192; kk += 4) {
      v2f a, b;
      a.x = ap[kk];
      a.y = ap[kk + 1];
      b.x = wp[kk * 128];
      b.y = wp[kk * 128 + 128];
      acc = __builtin_amdgcn_wmma_f32_16x16x4_f32(false, a, false, b,
                                                  (short)0, acc, false, false);
    }

    if (tid < 16 && scol2[cur][tid] >= 0) atomicAdd(&counts[scol2[cur][tid]], 1.0f);

    // ---- LayerNorm over 128 features (rows striped across the 8 waves)
    float h[8];
    #pragma unroll
    for (int v = 0; v < 8; ++v) h[v] = acc[v] + bias;
    #pragma unroll
    for (int v = 0; v < 8; ++v) {
      float s = h[v], q = h[v] * h[v];
      #pragma unroll
      for (int off = 1; off < 16; off <<= 1) {
        s += __shfl_xor(s, off, 32);
        q += __shfl_xor(q, off, 32);
      }
      if (l15 == 0) {
        int m = v + (hi << 3);
        red_s[wv * 16 + m] = s;
        red_q[wv * 16 + m] = q;
      }
    }
    __syncthreads();
    if (tid < 16) {
      float ts = 0.f, tq = 0.f;
      #pragma unroll
      for (int w = 0; w < 8; ++w) { ts += red_s[w * 16 + tid]; tq += red_q[w * 16 + tid]; }
      float mu  = ts * (1.0f / 128.0f);
      float var = tq * (1.0f / 128.0f) - mu * mu;
      smean[tid] = mu;
      srstd[tid] = rsqrtf(var + LN_EPS);
    }
    __syncthreads();

    // ---- LN affine + LeakyReLU + scatter-add
    #pragma unroll
    for (int v = 0; v < 8; ++v) {
      int m = v + (hi << 3);
      int col = scol2[cur][m];
      if (col >= 0) {
        float val = leaky((h[v] - smean[m]) * srstd[m] * gam + bet);
        atomicAdd(&sums[(long long)col * 128 + gn], val);
      }
    }
    __syncthreads();   // protect LDS buffers before next stage/compute round
  }
}

// ---------------------------------------------------------------- node MLP + residual
// Per 16-node tile: C(16x16) = Agg_raw(16x128) x W2(128x16) via 32 f32 WMMAs;
// the scatter-mean divide commutes through the GEMM (row scaling), so it is
// applied to the accumulator: h = sinv[m]*acc + b2.  Staging is a pure async copy.
__global__ __launch_bounds__(256) void k_node(
    const float* __restrict__ sums, const float* __restrict__ counts,
    const float* __restrict__ x, const float* __restrict__ W2,
    const float* __restrict__ b2, const float* __restrict__ g2,
    const float* __restrict__ be2, float* __restrict__ out, int N) {
  __shared__ float As[16 * 132];            // 16 nodes x 128 raw sums (pad->132)
  __shared__ float red_s[8 * 16];
  __shared__ float red_q[8 * 16];
  __shared__ float smean[16], srstd[16];
  __shared__ float sinv[16];

  const int tid  = threadIdx.x;
  const int lane = tid & 31;
  const int wv   = tid >> 5;
  const int l15  = lane & 15;
  const int hi   = lane >> 4;
  const int kh   = hi << 1;
  const int gn   = (wv << 4) + l15;
  const int nb0  = blockIdx.x * 16;

  // async-stage raw sums tile (2 b128 per thread)
  #pragma unroll
  for (int k = 0; k < 2; ++k) {
    int i = tid + k * 256;
    int m = i >> 5, c4 = i & 31;
    int node = nb0 + m; if (node >= N) node = N - 1;
    cp16_async(sums + (long long)node * 128 + c4 * 4, &As[m * 132 + c4 * 4]);
  }
  if (tid < 16) {
    int node = nb0 + tid;
    float cnt = (node < N) ? counts[node] : 1.0f;
    sinv[tid] = 1.0f / fmaxf(cnt, 1.0f);
  }
  wait_async_le<0>();
  __syncthreads();

  const float bias = b2[gn];
  const float gam  = g2[gn];
  const float bet  = be2[gn];
  const float* ap = &As[l15 * 132 + kh];
  const float* wp = W2 + kh * 128 + gn;

  v8f acc = {};
  #pragma unroll 8
  for (int kk = 0; kk < 128; kk += 4) {
    v2f a, b;
    a.x = ap[kk];
    a.y = ap[kk + 1];
    b.x = wp[kk * 128];
    b.y = wp[kk * 128 + 128];
    acc = __builtin_amdgcn_wmma_f32_16x16x4_f32(false, a, false, b,
                                                (short)0, acc, false, false);
  }

  float h[8];
  #pragma unroll
  for (int v = 0; v < 8; ++v) {
    int m = v + (hi << 3);
    h[v] = acc[v] * sinv[m] + bias;       // exact scatter-mean divide, folded
  }
  #pragma unroll
  for (int v = 0; v < 8; ++v) {
    float s = h[v], q = h[v] * h[v];
    #pragma unroll
    for (int off = 1; off < 16; off <<= 1) {
      s += __shfl_xor(s, off, 32);
      q += __shfl_xor(q, off, 32);
    }
    if (l15 == 0) {
      int m = v + (hi << 3);
      red_s[wv * 16 + m] = s;
      red_q[wv * 16 + m] = q;
    }
  }
  __syncthreads();
  if (tid < 16) {
    float ts = 0.f, tq = 0.f;
    #pragma unroll
    for (int w = 0; w < 8; ++w) { ts += red_s[w * 16 + tid]; tq += red_q[w * 16 + tid]; }
    float mu  = ts * (1.0f / 128.0f);
    float var = tq * (1.0f / 128.0f) - mu * mu;
    smean[tid] = mu;
    srstd[tid] = rsqrtf(var + LN_EPS);
  }
  __syncthreads();

  #pragma unroll
  for (int v = 0; v < 8; ++v) {
    int m = v + (hi << 3);
    int node = nb0 + m;
    if (node < N) {
      float val = leaky((h[v] - smean[m]) * srstd[m] * gam + bet);
      val = leaky(val + x[(long long)node * 128 + gn]);
      out[(long long)node * 128 + gn] = val;
    }
  }
}

// ---------------------------------------------------------------- launch
extern "C" void kernel_launch(void* const* d_in, const int* in_sizes, int n_in,
                              void* d_out, int out_size, void* d_ws, size_t ws_size,
                              hipStream_t stream) {
  const float*     x   = (const float*)d_in[0];
  const long long* ei  = (const long long*)d_in[1];
  const float*     ea  = (const float*)d_in[2];
  const float*     W1  = (const float*)d_in[3];
  const float*     b1  = (const float*)d_in[4];
  const float*     g1  = (const float*)d_in[5];
  const float*     be1 = (const float*)d_in[6];
  const float*     W2  = (const float*)d_in[7];
  const float*     b2  = (const float*)d_in[8];
  const float*     g2  = (const float*)d_in[9];
  const float*     be2 = (const float*)d_in[10];

  const int N = in_sizes[0] / 128;
  const int E = in_sizes[1] / 2;

  float* sums   = (float*)d_ws;                 // N x 128
  float* counts = sums + (size_t)N * 128;       // N

  int zn = N * 129;
  k_zero<<<(zn + 255) / 256, 256, 0, stream>>>(sums, zn);

  int tilesA = (E + 15) / 16;
  int gridA  = (tilesA + TILES_A - 1) / TILES_A;
  k_edge<<<gridA, 256, 0, stream>>>(x, ei, ea, W1, b1, g1, be1,
                                    sums, counts, E, (long long)E, tilesA);

  k_node<<<(N + 15) / 16, 256, 0, stream>>>(sums, counts, x, W2, b2, g2, be2,
                                            (float*)d_out, N);
}